// SubtractionNonCupy_56796647522323
// MI455X (gfx1250) — compile-verified
//
#include <hip/hip_runtime.h>
#include <hip/hip_bf16.h>

// ---------------------------------------------------------------------------
// SAN pairwise subtraction:  out[n,c,kk,pos] = x_center - x_neighbor(kk)
//   x: (8, 64, 56, 56) f32, k=7, s=1, p=3, d=1 -> out: (8, 64, 49, 3136) f32
//
// Strategy (MI455X):
//  * write-bandwidth bound (315 MB out vs 6.4 MB in) -> stage each input
//    plane ONCE in LDS via async global->LDS B128 copies (ASYNCcnt path),
//    zero-padded halo in LDS so the inner loop is branch-free.
//  * the subtraction is expressed as a rank-1 WMMA:
//       D(16x16) = A(16x4) * B(4x16) + C(16x16)
//    with A = e0 (col K=0 all ones), B row0 = centers, C = -patches.
//    One v_wmma_f32_16x16x4_f32 produces a 16(kk) x 16(pos) output tile.
// ---------------------------------------------------------------------------

typedef __attribute__((ext_vector_type(2))) float v2f;
typedef __attribute__((ext_vector_type(8))) float v8f;
typedef __attribute__((ext_vector_type(4))) int   v4i;

typedef __attribute__((address_space(1))) v4i gv4i;   // global int4
typedef __attribute__((address_space(3))) v4i lv4i;   // LDS int4

#define N_PLANES   (8 * 64)      // n*c
#define H          56
#define W          56
#define NPOS       (H * W)       // 3136
#define KK         49
#define LDS_STRIDE 64            // padded row stride (floats), 256B -> 16B aligned cols
#define LDS_ROWS   62            // 56 + 2*3 halo rows
#define LDS_ELEMS  (LDS_ROWS * LDS_STRIDE)
#define TILES      (NPOS / 16)   // 196 position tiles
#define CHUNKS     4             // grid.y, 49 tiles per chunk

static __device__ __forceinline__ void async_copy16(const float* g, float* l) {
    __builtin_amdgcn_global_load_async_to_lds_b128((gv4i*)g, (lv4i*)l, 0, 0);
}

static __device__ __forceinline__ void wait_async0() {
#if defined(__has_builtin) && __has_builtin(__builtin_amdgcn_s_wait_asynccnt)
    __builtin_amdgcn_s_wait_asynccnt(0);
#else
    asm volatile("s_wait_asynccnt 0" ::: "memory");
#endif
}

__global__ __launch_bounds__(256)
void san_sub_kernel(const float* __restrict__ x, float* __restrict__ out) {
    __shared__ float sm[LDS_ELEMS];

    const int nc   = blockIdx.x;          // plane (n*64 + c)
    const int q    = blockIdx.y;          // position chunk
    const int tid  = threadIdx.x;
    const int wave = tid >> 5;
    const int lane = tid & 31;

    // ---- zero LDS (halo needs zeros; interior is overwritten) -------------
    #pragma unroll 4
    for (int i = tid; i < LDS_ELEMS; i += 256) sm[i] = 0.0f;
    __syncthreads();

    // ---- async-stage the 56x56 plane into LDS interior --------------------
    // interior x[r][c] lives at sm[(r+3)*64 + 4 + c]; 16B-aligned B128 rows.
    const float* gplane = x + (size_t)nc * (H * W);
    for (int i = tid; i < (H * (W / 4)); i += 256) {   // 784 float4 chunks
        const int row = i / (W / 4);
        const int c4  = (i - row * (W / 4)) * 4;
        async_copy16(gplane + row * W + c4,
                     &sm[(row + 3) * LDS_STRIDE + 4 + c4]);
    }
    wait_async0();
    __syncthreads();

    // ---- compute: each wave owns 16-position tiles ------------------------
    float* outp = out + (size_t)nc * KK * NPOS;

    const int lo16 = lane & 15;           // column (position within tile)
    const int hi8  = (lane >> 4) << 3;    // 0 for lanes 0-15, 8 for 16-31

    // A matrix: only K=0 column nonzero (VGPR0, lanes 0-15) = 1.0
    v2f amat;
    amat.x = (lane < 16) ? 1.0f : 0.0f;
    amat.y = 0.0f;

    const int tile_end = (q + 1) * (TILES / CHUNKS);
    for (int t = q * (TILES / CHUNKS) + wave; t < tile_end; t += 8) {
        const int p = t * 16 + lo16;      // flat output position
        const int y = p / W;
        const int xcol = p - y * W;

        // center value for this column (padded coords: sm[(y+3)][x+4])
        const float cen = sm[(y + 3) * LDS_STRIDE + xcol + 4];

        // B matrix: only row K=0 nonzero (VGPR0, lanes 0-15) = centers
        v2f bmat;
        bmat.x = (lane < 16) ? cen : 0.0f;
        bmat.y = 0.0f;

        float* bp0 = outp + p;            // column base in output plane

        #pragma unroll
        for (int g = 0; g < 3; ++g) {     // kk groups 0-15, 16-31, 32-47
            v8f cmat;
            #pragma unroll
            for (int r = 0; r < 8; ++r) {
                const int kk = g * 16 + r + hi8;   // row M of the tile
                const int i  = kk / 7;
                const int j  = kk - i * 7;
                // patch = xp[y+i][x+j] -> sm[(y+i)*64 + (x+j+1)]; C = -patch
                cmat[r] = -sm[(y + i) * LDS_STRIDE + xcol + j + 1];
            }
            // D = A x B + C  ->  centers - patches  (16 kk-rows x 16 pos)
            v8f d = __builtin_amdgcn_wmma_f32_16x16x4_f32(
                false, amat, false, bmat, (short)0, cmat, false, false);

            float* bp = bp0 + (size_t)(g * 16 + hi8) * NPOS;
            #pragma unroll
            for (int r = 0; r < 8; ++r) bp[(size_t)r * NPOS] = d[r];
        }

        // kk = 48 (i=6, j=6) handled scalar on lanes 0-15
        const float v48 = cen - sm[(y + 6) * LDS_STRIDE + xcol + 7];
        if (lane < 16) outp[(size_t)48 * NPOS + p] = v48;
    }
}

extern "C" void kernel_launch(void* const* d_in, const int* in_sizes, int n_in,
                              void* d_out, int out_size, void* d_ws, size_t ws_size,
                              hipStream_t stream) {
    (void)in_sizes; (void)n_in; (void)out_size; (void)d_ws; (void)ws_size;
    const float* x = (const float*)d_in[0];
    float* out = (float*)d_out;
    dim3 grid(N_PLANES, CHUNKS);
    san_sub_kernel<<<grid, 256, 0, stream>>>(x, out);
}